// MambaMixer_31610959298744
// MI455X (gfx1250) — compile-verified
//
#include <hip/hip_runtime.h>

typedef __attribute__((ext_vector_type(2))) float v2f;
typedef __attribute__((ext_vector_type(8))) float v8f;

#define BM 128
#define BN 128
#define BK 32
#define LDSS 36                    // padded row stride (floats): 144B rows, 16B aligned, bank-spread
#define TILE_FLOATS (BM * LDSS)    // 4608 floats = 18432 B per tile buffer

// CDNA5 async global->LDS fill (GVS form: lds_vaddr, off_vaddr, saddr). Tracked by ASYNCcnt.
__device__ __forceinline__ void async_ld16(unsigned lds_off, unsigned byte_off, const float* base) {
    asm volatile("global_load_async_to_lds_b128 %0, %1, %2"
                 :: "v"(lds_off), "v"(byte_off), "s"(base)
                 : "memory");
}

template <int N>
__device__ __forceinline__ void wait_async() {
    asm volatile("s_wait_asynccnt %0" :: "n"(N) : "memory");
}

// NT GEMM: out[M,N] = A[M,K] * W[N,K]^T with mode-specific epilogue.
// MODE 0: n < 4096 -> out[m*4096+n] = v ; else out2[m*4096+n-4096] = silu(v)
// MODE 1: out[m*N+n] = v * Dv[n] * sz[m*N+n]
// MODE 2: out[m*N+n] = v
template <int MODE>
__global__ __launch_bounds__(256, 2) void wmma_gemm_nt(
    const float* __restrict__ A, const float* __restrict__ W,
    float* __restrict__ out, float* __restrict__ out2,
    const float* __restrict__ sz, const float* __restrict__ Dv,
    int M, int N, int K)
{
    __shared__ float As[2][TILE_FLOATS];
    __shared__ float Bs[2][TILE_FLOATS];

    const int tid  = threadIdx.x;
    const int wave = tid >> 5;      // 0..7
    const int lane = tid & 31;
    const int wm   = wave & 1;      // 2 waves along M (64 rows each)
    const int wn   = wave >> 1;     // 4 waves along N (32 cols each)
    const int tileM = blockIdx.x * BM;
    const int tileN = blockIdx.y * BN;

    const v8f zero8 = {0.f, 0.f, 0.f, 0.f, 0.f, 0.f, 0.f, 0.f};
    v8f acc[4][2];
#pragma unroll
    for (int i = 0; i < 4; ++i)
#pragma unroll
        for (int j = 0; j < 2; ++j) acc[i][j] = zero8;

    // cooperative fill mapping: 256 threads x float4, 4 row-passes per matrix
    const int lrow = tid >> 3;        // 0..31
    const int lcol = (tid & 7) * 4;   // 0,4,...,28

    const int ml = lane & 15;         // row/col within 16x16 fragment
    const int kb = (lane >> 4) * 2;   // K sub-offset for f32 16x16x4 A/B layout

    // LDS byte addresses = low 32 bits of generic pointers (ISA 10.2 LDS aperture)
    const unsigned ldsA0 = (unsigned)(unsigned long long)(const void*)&As[0][0];
    const unsigned ldsB0 = (unsigned)(unsigned long long)(const void*)&Bs[0][0];

    // issue one K-tile fill: 8 async b128 instructions per thread-wave (4 A + 4 B)
    auto issue = [&](int k0, int buf) {
        const unsigned bufoff = (unsigned)buf * (TILE_FLOATS * 4u);
#pragma unroll
        for (int rr = 0; rr < 4; ++rr) {
            const int r = lrow + rr * 32;
            const unsigned loff = bufoff + (unsigned)((r * LDSS + lcol) * 4);
            const unsigned aoff = (unsigned)(((size_t)(tileM + r) * K + k0 + lcol) * 4u);
            const unsigned boff = (unsigned)(((size_t)(tileN + r) * K + k0 + lcol) * 4u);
            async_ld16(ldsA0 + loff, aoff, A);
            async_ld16(ldsB0 + loff, boff, W);
        }
    };

    const int ntiles = K / BK;
    issue(0, 0);

    for (int t = 0; t < ntiles; ++t) {
        if (t + 1 < ntiles) {
            issue((t + 1) * BK, (t + 1) & 1);   // prefetch next tile into other buffer
            wait_async<8>();                    // this wave's tile-t fills (8 instrs) complete
        } else {
            wait_async<0>();
        }
        __syncthreads();                        // tile t visible block-wide

        const float* __restrict__ Ab = As[t & 1];
        const float* __restrict__ Bb = Bs[t & 1];
#pragma unroll
        for (int kk = 0; kk < BK; kk += 4) {
            v2f afrag[4], bfrag[2];
#pragma unroll
            for (int i = 0; i < 4; ++i)
                afrag[i] = *(const v2f*)(Ab + (wm * 64 + i * 16 + ml) * LDSS + kk + kb);
#pragma unroll
            for (int j = 0; j < 2; ++j)
                bfrag[j] = *(const v2f*)(Bb + (wn * 32 + j * 16 + ml) * LDSS + kk + kb);
#pragma unroll
            for (int i = 0; i < 4; ++i)
#pragma unroll
                for (int j = 0; j < 2; ++j)
                    acc[i][j] = __builtin_amdgcn_wmma_f32_16x16x4_f32(
                        false, afrag[i], false, bfrag[j],
                        (short)0, acc[i][j], false, false);
        }
        __syncthreads();                        // buffer t&1 drained; safe to refill next iter
    }

    // Epilogue. C/D layout: lane L, VGPR r -> m = r + (L>=16 ? 8 : 0), n = L%16
    const int mh = (lane >> 4) << 3;
#pragma unroll
    for (int i = 0; i < 4; ++i) {
#pragma unroll
        for (int j = 0; j < 2; ++j) {
            const int m0 = tileM + wm * 64 + i * 16 + mh;
            const int n  = tileN + wn * 32 + j * 16 + ml;
#pragma unroll
            for (int r = 0; r < 8; ++r) {
                float v = acc[i][j][r];
                const int m = m0 + r;
                if (MODE == 0) {
                    if (n < 4096) {
                        out[(size_t)m * 4096 + n] = v;
                    } else {
                        float s = v * (1.0f / (1.0f + __expf(-v)));  // silu(z)
                        out2[(size_t)m * 4096 + (n - 4096)] = s;
                    }
                } else if (MODE == 1) {
                    out[(size_t)m * N + n] = v * Dv[n] * sz[(size_t)m * N + n];
                } else {
                    out[(size_t)m * N + n] = v;
                }
            }
        }
    }
}

extern "C" void kernel_launch(void* const* d_in, const int* in_sizes, int n_in,
                              void* d_out, int out_size, void* d_ws, size_t ws_size,
                              hipStream_t stream) {
    // setup_inputs order: hidden_states, W_in, conv_w, W_x, W_dt, b_dt, D, W_out
    const float* hs     = (const float*)d_in[0];
    const float* W_in   = (const float*)d_in[1];
    const float* conv_w = (const float*)d_in[2];
    const float* Dv     = (const float*)d_in[6];
    const float* W_out  = (const float*)d_in[7];
    float* outp = (float*)d_out;

    const int M = 2 * 4096;                       // B_SZ * SEQ = 8192
    const size_t bufElems = (size_t)M * 4096;     // 128 MB per buffer
    float* xbuf  = (float*)d_ws;                  // x = hs @ W_in[:4096]^T
    float* szbuf = xbuf + bufElems;               // silu(hs @ W_in[4096:]^T)
    float* ubuf  = szbuf + bufElems;              // (x @ conv_w^T) * D * silu(z)

    dim3 blk(256);
    // GEMM1: [8192,2048] x [8192,2048]^T -> split x / silu(z)
    wmma_gemm_nt<0><<<dim3(M / BM, 8192 / BN), blk, 0, stream>>>(
        hs, W_in, xbuf, szbuf, nullptr, nullptr, M, 8192, 2048);
    // GEMM2: [8192,4096] x [4096,4096]^T, fused * D * silu(z)
    wmma_gemm_nt<1><<<dim3(M / BM, 4096 / BN), blk, 0, stream>>>(
        xbuf, conv_w, ubuf, nullptr, szbuf, Dv, M, 4096, 4096);
    // GEMM3: [8192,4096] x [2048,4096]^T -> final output
    wmma_gemm_nt<2><<<dim3(M / BM, 2048 / BN), blk, 0, stream>>>(
        ubuf, W_out, outp, nullptr, nullptr, nullptr, M, 2048, 4096);
}